// DeformConvNet_22900765623027
// MI455X (gfx1250) — compile-verified
//
#include <hip/hip_runtime.h>
#include <math.h>

typedef float v2f __attribute__((ext_vector_type(2)));
typedef float v8f __attribute__((ext_vector_type(8)));

#define BATCH 64

// ---------------------------------------------------------------------------
// conv1: Cin=1, Cout=32, 112x112, stride 1, pad 1, bias + ReLU (tiny FLOPs)
// ---------------------------------------------------------------------------
__global__ void conv1_bias_relu(const float* __restrict__ x, const float* __restrict__ w,
                                const float* __restrict__ bias, float* __restrict__ y) {
  const int HW = 112 * 112;
  long long idx = (long long)blockIdx.x * blockDim.x + threadIdx.x;
  if (idx >= (long long)BATCH * 32 * HW) return;
  int p = (int)(idx % HW);
  int t = (int)(idx / HW);
  int c = t & 31;
  int b = t >> 5;
  int i = p / 112, j = p % 112;
  const float* xs = x + (long long)b * HW;
  float s = bias[c];
#pragma unroll
  for (int r = 0; r < 3; ++r) {
    int iy = i + r - 1;
    if (iy < 0 || iy >= 112) continue;
#pragma unroll
    for (int q = 0; q < 3; ++q) {
      int ix = j + q - 1;
      if (ix < 0 || ix >= 112) continue;
      s += w[c * 9 + r * 3 + q] * xs[iy * 112 + ix];
    }
  }
  y[idx] = fmaxf(s, 0.0f);
}

// ---------------------------------------------------------------------------
// Weight transpose: (Cout, Cin, 3, 3) -> (tap, Cout, Cin), so the WMMA A
// operand is contiguous in Cin (one float2 per lane).
// ---------------------------------------------------------------------------
__global__ void transpose_w3x3(const float* __restrict__ w, float* __restrict__ wt,
                               int Cout, int Cin) {
  int idx = blockIdx.x * blockDim.x + threadIdx.x;
  int total = Cout * Cin * 9;
  if (idx >= total) return;
  int m = idx / (Cin * 9);
  int rem = idx % (Cin * 9);
  int ci = rem / 9;
  int tap = rem % 9;
  wt[((long long)tap * Cout + m) * Cin + ci] = w[idx];
}

// ---------------------------------------------------------------------------
// WMMA conv 3x3 (pad 1): per-tap GEMM, D = A(16xCin tile) x B(Cin x 16 pix),
// V_WMMA_F32_16X16X4_F32 (exact fp32). 4 waves / block cover 64 output chans;
// the shared B tile (Cin x 16 pixels) is staged in LDS via
// GLOBAL_LOAD_ASYNC_TO_LDS_B32, double-buffered across the 9 taps (the async
// engine prefetches tap t+1 while tap t feeds the WMMAs), eliminating the 4x
// redundant global reads across the block's waves.
// Requires: Cin % 8 == 0, Cout % 64 == 0, (Ho*Wo) % 16 == 0.
// ---------------------------------------------------------------------------
template <int CIN>
__global__ void __launch_bounds__(128) wmma_conv3x3(
    const float* __restrict__ in, const float* __restrict__ wt,
    const float* __restrict__ bias, float* __restrict__ out,
    int Cout, int H, int W, int Ho, int Wo, int stride, int relu) {
  constexpr int K = CIN / 8;  // async loads per thread per tap stage
  __shared__ float bstage[2][CIN * 16];

  int tid = threadIdx.x;
  int lane = tid & 31;
  int wave = tid >> 5;
  int half = lane >> 4;  // which K pair this lane holds in A/B operands
  int nl = lane & 15;    // M index (A) / N index (B)

  int b = blockIdx.z;
  int m0 = (blockIdx.y * 4 + wave) * 16;
  int n0 = blockIdx.x * 16;

  size_t HWs = (size_t)H * W;
  size_t CoutCin = (size_t)Cout * CIN;
  const float* inb = in + (size_t)b * CIN * HWs;

  // Staging role of this thread: fixed pixel column, 8-channel stride.
  int sn = tid & 15;
  int sci0 = tid >> 4;
  int snp = n0 + sn;
  int soy = snp / Wo, sox = snp % Wo;

  // Consumer role of this lane.
  int np = n0 + nl;
  int oy = np / Wo, ox = np % Wo;

  const float* abase = wt + (size_t)(m0 + nl) * CIN + half * 2;

  auto issue_stage = [&](int tap, int buf) {
    int r = tap / 3, q = tap % 3;
    int iy = soy * stride + r - 1;
    int ix = sox * stride + q - 1;
    bool v = (iy >= 0) && (iy < H) && (ix >= 0) && (ix < W);
    size_t poff = v ? ((size_t)iy * W + ix) : (size_t)0;  // clamped; zeroed on consume
    const float* gp = inb + (size_t)sci0 * HWs + poff;
    unsigned lp = (unsigned)(uintptr_t)(&bstage[buf][sci0 * 16 + sn]);
#pragma unroll
    for (int i = 0; i < K; ++i) {
      unsigned long long ga = (unsigned long long)(uintptr_t)gp;
      asm volatile("global_load_async_to_lds_b32 %0, %1, off"
                   :: "v"(lp), "v"(ga) : "memory");
      gp += (size_t)8 * HWs;  // 8 channels ahead
      lp += 8 * 16 * 4;       // 8 channels * 16 pixels * 4B in LDS
    }
  };

  v8f acc = {0.f, 0.f, 0.f, 0.f, 0.f, 0.f, 0.f, 0.f};

  issue_stage(0, 0);
  issue_stage(1, 1);

  for (int tap = 0; tap < 9; ++tap) {
    int cur = tap & 1;
    // Retire stage(tap): async loads complete in order, so <=K means the
    // older K (this tap) are done while stage(tap+1) may still be in flight.
    if (tap + 1 < 9)
      asm volatile("s_wait_asynccnt %0" :: "i"(K) : "memory");
    else
      asm volatile("s_wait_asynccnt 0" ::: "memory");
    __syncthreads();  // LDS writes from all waves' staging now visible

    int r = tap / 3, q = tap % 3;
    int iy = oy * stride + r - 1;
    int ix = ox * stride + q - 1;
    bool valid = (iy >= 0) && (iy < H) && (ix >= 0) && (ix < W);
    float mask = valid ? 1.0f : 0.0f;

    const float* ap = abase + (size_t)tap * CoutCin;
    const float* bsp = &bstage[cur][half * 32 + nl];
#pragma unroll 4
    for (int kk = 0; kk < CIN; kk += 4) {
      v2f a = *(const v2f*)ap;
      ap += 4;
      float b0 = bsp[0] * mask;
      float b1 = bsp[16] * mask;
      bsp += 64;
      v2f bv = {b0, b1};
      acc = __builtin_amdgcn_wmma_f32_16x16x4_f32(false, a, false, bv, 0, acc,
                                                  false, false);
    }
    __syncthreads();  // all waves done reading buf[cur]
    if (tap + 2 < 9) issue_stage(tap + 2, cur);  // refill the freed buffer
  }

  // C/D layout: VGPR v, lanes 0-15 -> M=v, lanes 16-31 -> M=v+8; N = lane&15.
  int mbase = m0 + half * 8;
  size_t HoWo = (size_t)Ho * Wo;
  float* op = out + ((size_t)b * Cout + mbase) * HoWo + np;
#pragma unroll
  for (int v = 0; v < 8; ++v) {
    float val = acc[v];
    if (bias) val += bias[mbase + v];
    if (relu) val = fmaxf(val, 0.0f);
    op[(size_t)v * HoWo] = val;
  }
}

// ---------------------------------------------------------------------------
// BatchNorm (training mode): deterministic two-stage reduction + apply.
// ---------------------------------------------------------------------------
__global__ void bn_stats(const float* __restrict__ y, float* __restrict__ psum,
                         float* __restrict__ psq, int C, int HW) {
  int c = blockIdx.x;
  int t = threadIdx.x;
  size_t total = (size_t)BATCH * HW;
  float s = 0.f, s2 = 0.f;
  for (size_t j = (size_t)blockIdx.y * 256 + t; j < total; j += (size_t)64 * 256) {
    size_t b = j / (size_t)HW;
    size_t p = j - b * (size_t)HW;
    float v = y[(b * C + c) * (size_t)HW + p];
    s += v;
    s2 += v * v;
  }
  __shared__ float sh[256], sh2[256];
  sh[t] = s; sh2[t] = s2;
  __syncthreads();
  for (int st = 128; st > 0; st >>= 1) {
    if (t < st) { sh[t] += sh[t + st]; sh2[t] += sh2[t + st]; }
    __syncthreads();
  }
  if (t == 0) {
    psum[c * 64 + blockIdx.y] = sh[0];
    psq[c * 64 + blockIdx.y] = sh2[0];
  }
}

__global__ void bn_finalize(const float* __restrict__ psum, const float* __restrict__ psq,
                            const float* __restrict__ g, const float* __restrict__ be,
                            float* __restrict__ ss, int HW) {
  __shared__ float sh[64], sh2[64];
  int c = blockIdx.x, t = threadIdx.x;
  sh[t] = psum[c * 64 + t];
  sh2[t] = psq[c * 64 + t];
  __syncthreads();
  for (int st = 32; st > 0; st >>= 1) {
    if (t < st) { sh[t] += sh[t + st]; sh2[t] += sh2[t + st]; }
    __syncthreads();
  }
  if (t == 0) {
    float cnt = (float)BATCH * (float)HW;
    float mean = sh[0] / cnt;
    float var = sh2[0] / cnt - mean * mean;
    float scale = g[c] * rsqrtf(var + 1e-5f);
    ss[2 * c] = scale;
    ss[2 * c + 1] = be[c] - mean * scale;
  }
}

__global__ void bn_apply(float* __restrict__ y, const float* __restrict__ ss,
                         int C, int HW, long long total) {
  long long idx = (long long)blockIdx.x * blockDim.x + threadIdx.x;
  if (idx >= total) return;
  int c = (int)((idx / HW) % C);
  y[idx] = y[idx] * ss[2 * c] + ss[2 * c + 1];
}

// ---------------------------------------------------------------------------
// Deformable bilinear resample. Offsets tensor (B,2C,H,W) viewed torch-style
// collapses to: per (b,c) slab of 2*H*W floats; off_i(p)=slab[2p], off_j=slab[2p+1].
// ---------------------------------------------------------------------------
__global__ void deform_resample(const float* __restrict__ x, const float* __restrict__ off,
                                float* __restrict__ out, int C, int H, int W,
                                long long total) {
  long long idx = (long long)blockIdx.x * blockDim.x + threadIdx.x;
  if (idx >= total) return;
  int HW = H * W;
  long long bc = idx / HW;
  int p = (int)(idx - bc * HW);
  int b = (int)(bc / C);
  int c = (int)(bc - (long long)b * C);
  const float* slab = off + ((size_t)b * 2 * C + 2 * c) * (size_t)HW;
  float oi = slab[2 * p];
  float oj = slab[2 * p + 1];
  float fi = (float)(p / W);
  float fj = (float)(p % W);
  float ci = fminf(fmaxf(oi + fi, 0.0f), (float)(H - 1));
  float cj = fminf(fmaxf(oj + fj, 0.0f), (float)(W - 1));
  float i0 = floorf(ci), i1 = ceilf(ci);
  float j0 = floorf(cj), j1 = ceilf(cj);
  int i0i = (int)i0, i1i = (int)i1, j0i = (int)j0, j1i = (int)j1;
  const float* xs = x + (size_t)bc * HW;
  float v00 = xs[i0i * W + j0i];
  float v10 = xs[i1i * W + j0i];
  float v01 = xs[i0i * W + j1i];
  float v11 = xs[i1i * W + j1i];
  float di = ci - i0, dj = cj - j0;
  float top = v00 + di * (v10 - v00);
  float bot = v01 + di * (v11 - v01);
  out[idx] = top + dj * (bot - top);
}

// ---------------------------------------------------------------------------
// Global average pool (64,128,28,28) -> (64,128)
// ---------------------------------------------------------------------------
__global__ void gap_pool(const float* __restrict__ y, float* __restrict__ pooled) {
  int bc = blockIdx.x;
  int t = threadIdx.x;
  const float* ys = y + (size_t)bc * 784;
  float s = 0.f;
  for (int i = t; i < 784; i += 256) s += ys[i];
  __shared__ float sh[256];
  sh[t] = s;
  __syncthreads();
  for (int st = 128; st > 0; st >>= 1) {
    if (t < st) sh[t] += sh[t + st];
    __syncthreads();
  }
  if (t == 0) pooled[bc] = sh[0] * (1.0f / 784.0f);
}

// ---------------------------------------------------------------------------
// FC (128->10) + softmax, one thread per batch row.
// ---------------------------------------------------------------------------
__global__ void fc_softmax(const float* __restrict__ pooled, const float* __restrict__ wfc,
                           const float* __restrict__ bfc, float* __restrict__ out) {
  int b = threadIdx.x;
  if (b >= BATCH) return;
  float acc[10];
#pragma unroll
  for (int o = 0; o < 10; ++o) acc[o] = bfc[o];
  for (int k = 0; k < 128; ++k) {
    float xv = pooled[b * 128 + k];
#pragma unroll
    for (int o = 0; o < 10; ++o) acc[o] += xv * wfc[o * 128 + k];
  }
  float m = acc[0];
#pragma unroll
  for (int o = 1; o < 10; ++o) m = fmaxf(m, acc[o]);
  float s = 0.f;
#pragma unroll
  for (int o = 0; o < 10; ++o) { acc[o] = expf(acc[o] - m); s += acc[o]; }
  float inv = 1.0f / s;
#pragma unroll
  for (int o = 0; o < 10; ++o) out[b * 10 + o] = acc[o] * inv;
}

// ---------------------------------------------------------------------------
// Host-side helpers
// ---------------------------------------------------------------------------
static void run_bn(float* y, const float* g, const float* be, int C, int HW,
                   float* psum, float* psq, float* ss, hipStream_t stream) {
  bn_stats<<<dim3(C, 64), 256, 0, stream>>>(y, psum, psq, C, HW);
  bn_finalize<<<C, 64, 0, stream>>>(psum, psq, g, be, ss, HW);
  long long total = (long long)BATCH * C * HW;
  bn_apply<<<(unsigned)((total + 255) / 256), 256, 0, stream>>>(y, ss, C, HW, total);
}

static void run_wmma_conv(const float* in, const float* wt, const float* bias, float* out,
                          int Cin, int Cout, int H, int W, int stride, int relu,
                          hipStream_t stream) {
  int Ho = (stride == 2) ? H / 2 : H;
  int Wo = (stride == 2) ? W / 2 : W;
  dim3 grid((unsigned)((Ho * Wo) / 16), (unsigned)(Cout / 64), BATCH);
  if (Cin == 32) {
    wmma_conv3x3<32><<<grid, 128, 0, stream>>>(in, wt, bias, out, Cout, H, W, Ho, Wo,
                                               stride, relu);
  } else if (Cin == 64) {
    wmma_conv3x3<64><<<grid, 128, 0, stream>>>(in, wt, bias, out, Cout, H, W, Ho, Wo,
                                               stride, relu);
  } else {
    wmma_conv3x3<128><<<grid, 128, 0, stream>>>(in, wt, bias, out, Cout, H, W, Ho, Wo,
                                                stride, relu);
  }
}

static void run_resample(const float* x, const float* off, float* out, int C, int H, int W,
                         hipStream_t stream) {
  long long total = (long long)BATCH * C * H * W;
  deform_resample<<<(unsigned)((total + 255) / 256), 256, 0, stream>>>(x, off, out, C, H, W,
                                                                      total);
}

extern "C" void kernel_launch(void* const* d_in, const int* in_sizes, int n_in,
                              void* d_out, int out_size, void* d_ws, size_t ws_size,
                              hipStream_t stream) {
  (void)in_sizes; (void)n_in; (void)out_size; (void)ws_size;

  const float* x     = (const float*)d_in[0];
  const float* w11   = (const float*)d_in[1];
  const float* b11   = (const float*)d_in[2];
  const float* g11   = (const float*)d_in[3];
  const float* be11  = (const float*)d_in[4];
  const float* woff12 = (const float*)d_in[5];
  const float* w12   = (const float*)d_in[6];
  const float* b12   = (const float*)d_in[7];
  const float* g12   = (const float*)d_in[8];
  const float* be12  = (const float*)d_in[9];
  const float* woff21 = (const float*)d_in[10];
  const float* w21   = (const float*)d_in[11];
  const float* b21   = (const float*)d_in[12];
  const float* g21   = (const float*)d_in[13];
  const float* be21  = (const float*)d_in[14];
  const float* woff22 = (const float*)d_in[15];
  const float* w22   = (const float*)d_in[16];
  const float* b22   = (const float*)d_in[17];
  const float* g22   = (const float*)d_in[18];
  const float* be22  = (const float*)d_in[19];
  const float* wfc   = (const float*)d_in[20];
  const float* bfc   = (const float*)d_in[21];
  float* out = (float*)d_out;

  // Workspace layout (bytes): buffers reused across stages; peak ~465 MB.
  char* ws = (char*)d_ws;
  float* R0 = (float*)(ws + 0ull);          // 102,760,448 B : y1 / y3
  float* R1 = (float*)(ws + 102760448ull);  // 205,520,896 B : off12 / off21 / off22
  float* R2 = (float*)(ws + 308281344ull);  // 102,760,448 B : d1 / d2 / d3
  float* R3 = (float*)(ws + 411041792ull);  //  51,380,224 B : y2 / y4
  float* misc = (float*)(ws + 462422016ull);
  float* wt_off12 = misc;                   // 9*64*32   = 18432
  float* wt_12    = wt_off12 + 18432;       // 18432
  float* wt_off21 = wt_12 + 18432;          // 9*128*64  = 73728
  float* wt_21    = wt_off21 + 73728;       // 73728
  float* wt_off22 = wt_21 + 73728;          // 9*256*128 = 294912
  float* wt_22    = wt_off22 + 294912;      // 9*128*128 = 147456
  float* psum     = wt_22 + 147456;         // 128*64
  float* psq      = psum + 8192;            // 128*64
  float* ssbuf    = psq + 8192;             // 2*128
  float* pooled   = ssbuf + 256;            // 64*128

  // Pre-transpose all GEMM weights to (tap, Cout, Cin).
  transpose_w3x3<<<72, 256, 0, stream>>>(woff12, wt_off12, 64, 32);
  transpose_w3x3<<<72, 256, 0, stream>>>(w12, wt_12, 64, 32);
  transpose_w3x3<<<288, 256, 0, stream>>>(woff21, wt_off21, 128, 64);
  transpose_w3x3<<<288, 256, 0, stream>>>(w21, wt_21, 128, 64);
  transpose_w3x3<<<1152, 256, 0, stream>>>(woff22, wt_off22, 256, 128);
  transpose_w3x3<<<576, 256, 0, stream>>>(w22, wt_22, 128, 128);

  // Stage 1: conv1 + ReLU + BN -> y1 (64,32,112,112) in R0
  {
    long long total = (long long)BATCH * 32 * 112 * 112;
    conv1_bias_relu<<<(unsigned)((total + 255) / 256), 256, 0, stream>>>(x, w11, b11, R0);
    run_bn(R0, g11, be11, 32, 112 * 112, psum, psq, ssbuf, stream);
  }
  // deform 1: off12 = conv(y1) (64ch), resample -> d1 in R2
  run_wmma_conv(R0, wt_off12, nullptr, R1, 32, 64, 112, 112, 1, 0, stream);
  run_resample(R0, R1, R2, 32, 112, 112, stream);
  // Stage 2: conv12 stride2 + ReLU + BN -> y2 (64,64,56,56) in R3
  run_wmma_conv(R2, wt_12, b12, R3, 32, 64, 112, 112, 2, 1, stream);
  run_bn(R3, g12, be12, 64, 56 * 56, psum, psq, ssbuf, stream);
  // deform 2: off21 (128ch) in R1, resample -> d2 in R2
  run_wmma_conv(R3, wt_off21, nullptr, R1, 64, 128, 56, 56, 1, 0, stream);
  run_resample(R3, R1, R2, 64, 56, 56, stream);
  // Stage 3: conv21 stride1 + ReLU + BN -> y3 (64,128,56,56) in R0
  run_wmma_conv(R2, wt_21, b21, R0, 64, 128, 56, 56, 1, 1, stream);
  run_bn(R0, g21, be21, 128, 56 * 56, psum, psq, ssbuf, stream);
  // deform 3: off22 (256ch) in R1, resample -> d3 in R2
  run_wmma_conv(R0, wt_off22, nullptr, R1, 128, 256, 56, 56, 1, 0, stream);
  run_resample(R0, R1, R2, 128, 56, 56, stream);
  // Stage 4: conv22 stride2 + ReLU + BN -> y4 (64,128,28,28) in R3
  run_wmma_conv(R2, wt_22, b22, R3, 128, 128, 56, 56, 2, 1, stream);
  run_bn(R3, g22, be22, 128, 28 * 28, psum, psq, ssbuf, stream);
  // Pool + FC + softmax
  gap_pool<<<BATCH * 128, 256, 0, stream>>>(R3, pooled);
  fc_softmax<<<1, 64, 0, stream>>>(pooled, wfc, bfc, out);
}